// DictLearnEMA_67963562491996
// MI455X (gfx1250) — compile-verified
//
#include <hip/hip_runtime.h>

typedef __attribute__((ext_vector_type(2))) float v2f;
typedef __attribute__((ext_vector_type(8))) float v8f;

#define N_TOK 32768
#define DDIM  128
#define KDIM  1024
#define NUMEL 4194304   // B*D*H*W = N_TOK*DDIM
#define REP_OUT_OFF 4194306L  // 1 (loss) + NUMEL (recon_out) + 1 (perplexity)
#define PERP_OUT_OFF 4194305L

// ---- workspace layout (float offsets) ----
static const long XF_OFF    = 0;                        // N*D        = 4,194,304
static const long RPRE_OFF  = 4194304;                  // N*K        = 33,554,432
static const long IDX_OFF   = RPRE_OFF + 33554432L;     // N*8 ints   = 262,144
static const long WNORM_OFF = IDX_OFF + 262144L;        // 1024
static const long CM_OFF    = WNORM_OFF + 1024L;        // 1024
static const long CA_OFF    = CM_OFF + 1024L;           // 1024
static const long CB_OFF    = CA_OFF + 1024L;           // 1024
static const long PART_OFF  = CB_OFF + 1024L;           // 8*1024*3   = 24,576
static const long CNT_OFF   = PART_OFF + 24576L;        // 1024 ints
static const long LOSS_OFF  = CNT_OFF + 1024L;          // 8192 per-block partials

// ---------------------------------------------------------------------------
// prep: wnorm[k] = ||dict_w[k]||^2 ; zero usage counts
__global__ __launch_bounds__(256) void dle_prep(const float* __restrict__ dictw,
                                                float* __restrict__ wnorm,
                                                int* __restrict__ counts) {
    int k = blockIdx.x * 256 + threadIdx.x;   // grid 4x256 -> 1024
    const float* row = dictw + (long)k * DDIM;
    float s = 0.f;
    for (int d = 0; d < DDIM; ++d) s += row[d] * row[d];
    wnorm[k]  = s;
    counts[k] = 0;
}

// ---------------------------------------------------------------------------
// transpose x [B,D,HW] -> xf [B*HW, D] so GEMM A-tiles are contiguous rows
__global__ __launch_bounds__(256) void dle_transpose(const float* __restrict__ x,
                                                     float* __restrict__ xf) {
    __shared__ float tile[32][33];
    int b  = blockIdx.z;
    int d0 = blockIdx.y * 32;
    int h0 = blockIdx.x * 32;
    int tx = threadIdx.x & 31, ty = threadIdx.x >> 5;   // ty 0..7
    for (int i = 0; i < 32; i += 8)
        tile[ty + i][tx] = x[((long)(b * 128 + d0 + ty + i)) * 1024 + h0 + tx];
    __syncthreads();
    for (int i = 0; i < 32; i += 8)
        xf[((long)(b * 1024 + h0 + ty + i)) * 128 + d0 + tx] = tile[tx][ty + i];
}

// ---------------------------------------------------------------------------
// Fused dual f32 WMMA GEMM (r_pre and distance score) + in-LDS top-8 per row.
// One block = 16-row strip x full K=1024. 8 waves, each wave: 8 16x16 tiles.
__global__ __launch_bounds__(256) void dle_gemm_topk(const float* __restrict__ xf,
                                                     const float* __restrict__ linw,
                                                     const float* __restrict__ linb,
                                                     const float* __restrict__ dictw,
                                                     const float* __restrict__ wnorm,
                                                     float* __restrict__ rpre,
                                                     int* __restrict__ topidx) {
    __shared__ float sA[16 * 132];     // A tile 16x128, pad 4 -> conflict-free col reads
    __shared__ float sS[16 * 1040];    // score strip 16x1024 (padded)
    const int tid   = threadIdx.x;
    const int nbase = blockIdx.x * 16;

    // stage A tile (coalesced)
    for (int i = 0; i < 8; ++i) {
        int e = tid + 256 * i;
        int r = e >> 7, c = e & 127;
        sA[r * 132 + c] = xf[(long)(nbase + r) * DDIM + c];
    }
    __syncthreads();

    const int lane = tid & 31;
    const int wave = tid >> 5;
    const int m16  = lane & 15;
    const int koff = (lane < 16) ? 0 : 2;   // ISA 16x4 f32 A/B lane layout

    for (int tt = 0; tt < 8; ++tt) {
        const int jt = wave * 128 + tt * 16;           // column-tile base
        v8f accR = {};                                  // r_pre accumulator
        v8f accW = {};                                  // x.dict_w^T accumulator
        const float* bL = linw  + (long)(jt + m16) * DDIM + koff;
        const float* bD = dictw + (long)(jt + m16) * DDIM + koff;
        const int aoff  = m16 * 132 + koff;
        for (int kk = 0; kk < DDIM; kk += 4) {
            v2f a;
            a[0] = sA[aoff + kk];
            a[1] = sA[aoff + kk + 1];
            v2f bl = *(const v2f*)(bL + kk);
            v2f bd = *(const v2f*)(bD + kk);
            accR = __builtin_amdgcn_wmma_f32_16x16x4_f32(false, a, false, bl,
                                                         (short)0, accR, false, false);
            accW = __builtin_amdgcn_wmma_f32_16x16x4_f32(false, a, false, bd,
                                                         (short)0, accW, false, false);
        }
        // C/D layout: VGPR i -> M=i (lanes 0-15) / M=i+8 (lanes 16-31); N = lane%16
        const int col = jt + m16;
        const float wb = wnorm[col];
        const float lb = linb[col];
        const int mro  = (lane < 16) ? 0 : 8;
        for (int i = 0; i < 8; ++i) {
            int mr = i + mro;
            rpre[(long)(nbase + mr) * KDIM + col] = accR[i] + lb;
            sS[mr * 1040 + col] = 2.0f * accW[i] - wb;   // == d + ||x_n||^2 (rank-safe)
        }
    }
    __syncthreads();

    // top-8 per row: iterative argmax with removal (tie -> lower index, matches top_k)
    for (int rr = 0; rr < 2; ++rr) {
        const int row = wave * 2 + rr;
        const int gn  = nbase + row;
        for (int it = 0; it < 8; ++it) {
            float bv = -3.0e38f; int bc = -1;
            for (int c = lane; c < KDIM; c += 32) {
                float v = sS[row * 1040 + c];
                if (v > bv || (v == bv && c < bc)) { bv = v; bc = c; }
            }
            for (int off = 16; off > 0; off >>= 1) {
                float ov = __shfl_xor(bv, off, 32);
                int   oc = __shfl_xor(bc, off, 32);
                if (ov > bv || (ov == bv && oc < bc)) { bv = ov; bc = oc; }
            }
            if (lane == 0) {
                topidx[(long)gn * 8 + it] = bc;
                sS[row * 1040 + bc] = -3.0e38f;
            }
            __syncthreads();   // uniform trip counts across all 8 waves
        }
    }
}

// ---------------------------------------------------------------------------
// Online column stats over r_pre: per column max m, S1=sum e^(v-m), S2=sum e^(2(v-m))
__global__ __launch_bounds__(256) void dle_colstats(const float* __restrict__ rpre,
                                                    float* __restrict__ partial) {
    const int t     = threadIdx.x;
    const int c     = blockIdx.x * 64 + (t & 63);
    const int rl    = t >> 6;                // 0..3 row-split within block
    const int chunk = blockIdx.y;            // 0..7 row chunks of 4096
    float m = -3.0e38f, s1 = 0.f, s2 = 0.f;
    const int n0 = chunk * 4096;
    for (int n = n0 + rl; n < n0 + 4096; n += 4) {
        float v = rpre[(long)n * KDIM + c];
        if (v > m) {
            float e = __expf(m - v);
            s1 = s1 * e + 1.0f;
            s2 = s2 * e * e + 1.0f;
            m = v;
        } else {
            float e = __expf(v - m);
            s1 += e;
            s2 += e * e;
        }
    }
    __shared__ float sm[256], sa[256], sb[256];
    sm[t] = m; sa[t] = s1; sb[t] = s2;
    __syncthreads();
    if (rl == 0) {   // t < 64
        for (int o = 1; o < 4; ++o) {
            float om = sm[t + 64 * o], o1 = sa[t + 64 * o], o2 = sb[t + 64 * o];
            float M  = fmaxf(m, om);
            float ea = __expf(m - M), eb = __expf(om - M);
            s1 = s1 * ea + o1 * eb;
            s2 = s2 * ea * ea + o2 * eb * eb;
            m  = M;
        }
        long pb = ((long)chunk * KDIM + c) * 3;
        partial[pb] = m; partial[pb + 1] = s1; partial[pb + 2] = s2;
    }
}

// ---------------------------------------------------------------------------
// Merge chunk partials; fold softmax+BatchNorm into per-column affine:
//   rhat(n,k) = exp(rpre - cM[k]) * cA[k] + cB[k]
// (softmax column mean is exactly 1/N; var = S2/(S1^2 N) - 1/N^2)
__global__ __launch_bounds__(256) void dle_colfinal(const float* __restrict__ partial,
                                                    const float* __restrict__ gamma,
                                                    const float* __restrict__ beta,
                                                    float* __restrict__ cM,
                                                    float* __restrict__ cA,
                                                    float* __restrict__ cB) {
    int k = blockIdx.x * 256 + threadIdx.x;  // grid 4x256
    float m = -3.0e38f, s1 = 0.f, s2 = 0.f;
    for (int ch = 0; ch < 8; ++ch) {
        long pb = ((long)ch * KDIM + k) * 3;
        float om = partial[pb], o1 = partial[pb + 1], o2 = partial[pb + 2];
        float M  = fmaxf(m, om);
        float ea = __expf(m - M), eb = __expf(om - M);
        s1 = s1 * ea + o1 * eb;
        s2 = s2 * ea * ea + o2 * eb * eb;
        m  = M;
    }
    const float invN  = 1.0f / (float)N_TOK;
    const float inv   = 1.0f / s1;
    const float var   = s2 * inv * inv * invN - invN * invN;
    const float scale = gamma[k] * rsqrtf(var + 1e-5f);
    cM[k] = m;
    cA[k] = inv * scale;
    cB[k] = beta[k] - scale * invN;
}

// ---------------------------------------------------------------------------
// Scatter rep (dense write, 8 nnz/row), sparse recon, straight-through out,
// loss partials (deterministic), usage counts. One block = 4 token rows.
__global__ __launch_bounds__(256) void dle_scatter(const float* __restrict__ rpre,
                                                   const int* __restrict__ topidx,
                                                   const float* __restrict__ cM,
                                                   const float* __restrict__ cA,
                                                   const float* __restrict__ cB,
                                                   const float* __restrict__ dictw,
                                                   const float* __restrict__ xflat,
                                                   float* __restrict__ out,
                                                   int* __restrict__ counts,
                                                   float* __restrict__ lossPart) {
    __shared__ int   sIdx[4][8];
    __shared__ float sVal[4][8];
    __shared__ float sRed[256];
    const int t  = threadIdx.x;
    const int nb = blockIdx.x * 4;

    if (t < 32) {
        int r = t >> 3, j = t & 7;
        int n = nb + r;
        int k = topidx[(long)n * 8 + j];
        sIdx[r][j] = k;
        float e = __expf(rpre[(long)n * KDIM + k] - cM[k]);
        sVal[r][j] = e * cA[k] + cB[k];
        atomicAdd(&counts[k], 1);
    }
    __syncthreads();

    // dense rep write (zeros except top-8 columns); base only 8B-aligned -> float2
    float* rep = out + REP_OUT_OFF;
    {
        const int r     = t >> 6;            // 0..3
        const int cbase = (t & 63) * 16;
        const long rowb = (long)(nb + r) * KDIM;
        for (int ch = 0; ch < 8; ++ch) {
            int c0 = cbase + ch * 2;
            float2 v = make_float2(0.f, 0.f);
            for (int j = 0; j < 8; ++j) {
                int dlt = sIdx[r][j] - c0;
                if ((unsigned)dlt < 2u) {
                    if (dlt == 0) v.x = sVal[r][j]; else v.y = sVal[r][j];
                }
            }
            *(float2*)(rep + rowb + c0) = v;
        }
    }

    // recon = sum_j val_j * dict_w[idx_j,:]; raw-view pairing with x (flat index)
    float lloss = 0.f;
    const int r2 = t >> 7;       // 0..1
    const int d  = t & 127;
    for (int rr = r2; rr < 4; rr += 2) {
        float acc = 0.f;
        for (int j = 0; j < 8; ++j)
            acc += sVal[rr][j] * dictw[(long)sIdx[rr][j] * DDIM + d];
        long fi  = (long)(nb + rr) * DDIM + d;
        float xv = xflat[fi];
        out[1 + fi] = 2.0f * acc - xv;       // recon + sg(recon - x)
        float dd = xv - acc;
        lloss += dd * dd;
    }
    sRed[t] = lloss;
    __syncthreads();
    for (int o = 128; o > 0; o >>= 1) {
        if (t < o) sRed[t] += sRed[t + o];
        __syncthreads();
    }
    if (t == 0) lossPart[blockIdx.x] = sRed[0];
}

// ---------------------------------------------------------------------------
// finalize loss and perplexity
__global__ __launch_bounds__(1024) void dle_final(const int* __restrict__ counts,
                                                  const float* __restrict__ lossPart,
                                                  const float* __restrict__ cc,
                                                  float* __restrict__ out) {
    __shared__ float sH[1024];
    __shared__ float sL[1024];
    const int t = threadIdx.x;
    float avg = (float)counts[t] / (8.0f * (float)N_TOK);
    sH[t] = -avg * __logf(avg + 1e-8f);
    float ls = 0.f;
    for (int i = t; i < 8192; i += 1024) ls += lossPart[i];
    sL[t] = ls;
    __syncthreads();
    for (int o = 512; o > 0; o >>= 1) {
        if (t < o) { sH[t] += sH[t + o]; sL[t] += sL[t + o]; }
        __syncthreads();
    }
    if (t == 0) {
        out[PERP_OUT_OFF] = __expf(sH[0]);
        out[0] = sL[0] * (cc[0] + 1.0f) / (float)NUMEL;
    }
}

// ---------------------------------------------------------------------------
extern "C" void kernel_launch(void* const* d_in, const int* in_sizes, int n_in,
                              void* d_out, int out_size, void* d_ws, size_t ws_size,
                              hipStream_t stream) {
    const float* x      = (const float*)d_in[0];
    const float* lin_w  = (const float*)d_in[1];
    const float* lin_b  = (const float*)d_in[2];
    const float* gamma  = (const float*)d_in[3];
    const float* beta   = (const float*)d_in[4];
    const float* dict_w = (const float*)d_in[5];
    const float* cc     = (const float*)d_in[6];
    float* out = (float*)d_out;
    float* ws  = (float*)d_ws;

    float* xf     = ws + XF_OFF;
    float* rpre   = ws + RPRE_OFF;
    int*   topidx = (int*)(ws + IDX_OFF);
    float* wnorm  = ws + WNORM_OFF;
    float* cMp    = ws + CM_OFF;
    float* cAp    = ws + CA_OFF;
    float* cBp    = ws + CB_OFF;
    float* part   = ws + PART_OFF;
    int*   counts = (int*)(ws + CNT_OFF);
    float* lossP  = ws + LOSS_OFF;

    dle_prep<<<4, 256, 0, stream>>>(dict_w, wnorm, counts);
    dle_transpose<<<dim3(32, 4, 32), 256, 0, stream>>>(x, xf);
    dle_gemm_topk<<<N_TOK / 16, 256, 0, stream>>>(xf, lin_w, lin_b, dict_w, wnorm,
                                                  rpre, topidx);
    dle_colstats<<<dim3(16, 8), 256, 0, stream>>>(rpre, part);
    dle_colfinal<<<4, 256, 0, stream>>>(part, gamma, beta, cMp, cAp, cBp);
    dle_scatter<<<N_TOK / 4, 256, 0, stream>>>(rpre, topidx, cMp, cAp, cBp, dict_w,
                                               x, out, counts, lossP);
    dle_final<<<1, 1024, 0, stream>>>(counts, lossP, cc, out);
}